// ASTEncoder_34093450396344
// MI455X (gfx1250) — compile-verified
//
#include <hip/hip_runtime.h>

// TreeLSTM (complete binary heap) level-parallel implementation for gfx1250.
// Each level d (9..0) is one batched GEMM (64*2^d x 768) @ (768 x 2048)
// computed with V_WMMA_F32_16X16X4_F32 (exact fp32), with the LSTM-cell
// activations and the child->parent pair reduction fused into the epilogue.
// Workgroup tile: 32 rows x (4 gates x 32 cols); each of the 8 waves owns two
// stacked 16x16 WMMA tiles (M-blocking x2) so every B fragment feeds 2 WMMAs.

typedef float v2f __attribute__((ext_vector_type(2)));
typedef float v8f __attribute__((ext_vector_type(8)));

#define B_DIM   64
#define N_NODES 1023
#define E_DIM   256
#define H_DIM   512
#define K_DIM   768   // E + H

__device__ __forceinline__ float sigm(float x) { return 1.0f / (1.0f + __expf(-x)); }

__device__ __forceinline__ v8f wmma_f32(v2f a, v2f b, v8f c) {
  return __builtin_amdgcn_wmma_f32_16x16x4_f32(false, a, false, b,
                                               (short)0, c, false, false);
}

// Inner K-loop with compile-time B stride so k0*NCOLS folds into the
// instruction's 24-bit immediate offset (no per-load 64-bit address VALU).
template <int NCOLS>
__device__ __forceinline__ void gemm_k(const float* __restrict__ pB,
                                       const float* __restrict__ pA0,
                                       const float* __restrict__ pA1,
                                       v8f& acc0, v8f& acc1) {
  #pragma unroll 8
  for (int k0 = 0; k0 < K_DIM; k0 += 4) {
    v2f b;
    b.x = pB[(size_t)k0 * NCOLS];
    b.y = pB[(size_t)(k0 + 1) * NCOLS];
    v2f a0 = *(const v2f*)(pA0 + k0);
    v2f a1 = *(const v2f*)(pA1 + k0);
    acc0 = wmma_f32(a0, b, acc0);
    acc1 = wmma_f32(a1, b, acc1);
  }
}

__global__ void __launch_bounds__(256)
tree_lstm_level(const int*   __restrict__ tok,    // ast_nodes (B, N) int32
                const float* __restrict__ emb,    // (VOCAB, 256)
                const float* __restrict__ Wiou,   // (768, 1536)
                const float* __restrict__ biou,   // (1536)
                const float* __restrict__ Wf,     // (768, 512)
                const float* __restrict__ bf,     // (512)
                const float* __restrict__ Hin,    // hsum for this level (rows x 512) or null
                const float* __restrict__ Cin,    // csum for this level or null
                float*       __restrict__ Hout,   // hsum for parent level (rows/2 x 512)
                float*       __restrict__ Cout,   // csum for parent level
                float*       __restrict__ outRoot,// (64, 512) final h_root
                int L, int log2L, int n0, int hasState)
{
  // LDS: A tile 32x768 (pitch 772 -> conflict-free WMMA A reads, float4-aligned)
  __shared__ float sA[32][772];        // 98,816 B
  __shared__ float sGate[4][32][33];   // i,o,u,f preact tiles 32x32 (16,896 B)
  __shared__ float sH[32][33];
  __shared__ float sC[32][33];
  __shared__ int   sTok[32];

  const int tid  = threadIdx.x;
  const int lane = tid & 31;
  const int w    = tid >> 5;          // wave 0..7
  const int m15  = lane & 15;
  const int hi   = lane >> 4;         // K-half selector within WMMA fragment
  const int r0   = blockIdx.x * 32;   // row tile (row = b*L + j)
  const int t0   = blockIdx.y * 32;   // gate-column tile

  // ---- token gather for the 32 rows of this tile ----
  if (tid < 32) {
    int grow = r0 + tid;
    int b = grow >> log2L;
    int j = grow & (L - 1);
    sTok[tid] = tok[b * N_NODES + n0 + j];
  }
  __syncthreads();

  // ---- stage A = [emb_row | hsum_row] into LDS ----
  {
    int m  = tid >> 3;   // 0..31 row
    int kc = tid & 7;    // float4 column phase (8 per row)
    const float4* embRow = (const float4*)(emb + (size_t)sTok[m] * E_DIM);
    #pragma unroll
    for (int it = 0; it < 8; ++it) {           // k = 0..255
      float4 v = embRow[kc + 8 * it];
      *(float4*)&sA[m][(kc + 8 * it) * 4] = v;
    }
    if (hasState) {
      const float4* hRow = (const float4*)(Hin + (size_t)(r0 + m) * H_DIM);
      #pragma unroll
      for (int it = 8; it < 24; ++it) {        // k = 256..767
        float4 v = hRow[kc + 8 * it - 64];
        *(float4*)&sA[m][(kc + 8 * it) * 4] = v;
      }
    } else {
      float4 z = make_float4(0.f, 0.f, 0.f, 0.f);
      #pragma unroll
      for (int it = 8; it < 24; ++it)
        *(float4*)&sA[m][(kc + 8 * it) * 4] = z;
    }
  }
  __syncthreads();

  // ---- per-wave: two stacked 16x16 fp32 WMMA tiles (gate g, col half nh) ----
  const int g  = w >> 1;           // 0:i 1:o 2:u 3:f
  const int nh = (w & 1) * 16;
  // A fragments: elem0 = A[m][k0+2*hi], elem1 = A[m][k0+2*hi+1]
  const float* pA0 = &sA[m15][2 * hi];
  const float* pA1 = &sA[m15 + 16][2 * hi];

  v8f acc0 = {0.f, 0.f, 0.f, 0.f, 0.f, 0.f, 0.f, 0.f};
  v8f acc1 = acc0;
  if (g < 3) {
    const int colg = g * H_DIM + t0 + nh + m15;
    const float* pB = Wiou + (size_t)(2 * hi) * (3 * H_DIM) + colg;
    gemm_k<3 * H_DIM>(pB, pA0, pA1, acc0, acc1);
  } else {
    const int colg = t0 + nh + m15;
    const float* pB = Wf + (size_t)(2 * hi) * H_DIM + colg;
    gemm_k<H_DIM>(pB, pA0, pA1, acc0, acc1);
  }

  // D layout: acc[r] -> row (r + 8*hi), col (lane&15)
  #pragma unroll
  for (int r = 0; r < 8; ++r) {
    sGate[g][r + 8 * hi][nh + m15]      = acc0[r];
    sGate[g][16 + r + 8 * hi][nh + m15] = acc1[r];
  }
  __syncthreads();

  // ---- fused LSTM cell: c = sig(i)*tanh(u) + sig(f)*csum ; h = sig(o)*tanh(c)
  #pragma unroll
  for (int e = tid; e < 1024; e += 256) {
    int m  = e >> 5;
    int cc = e & 31;
    int col = t0 + cc;
    float iv = sGate[0][m][cc] + biou[col];
    float ov = sGate[1][m][cc] + biou[H_DIM + col];
    float uv = sGate[2][m][cc] + biou[2 * H_DIM + col];
    float fv = sGate[3][m][cc] + bf[col];
    float cs = hasState ? Cin[(size_t)(r0 + m) * H_DIM + col] : 0.f;
    float c  = sigm(iv) * tanhf(uv) + sigm(fv) * cs;
    float h  = sigm(ov) * tanhf(c);
    sH[m][cc] = h;
    sC[m][cc] = c;
  }
  __syncthreads();

  if (L == 1) {
    // root: rows are batch entries, write h directly to output
    #pragma unroll
    for (int e = tid; e < 1024; e += 256) {
      int m = e >> 5, cc = e & 31;
      outRoot[(size_t)(r0 + m) * H_DIM + t0 + cc] = sH[m][cc];
    }
  } else {
    // deterministic child-pair reduction into parent hsum/csum.
    // children (2p, 2p+1) are adjacent even-aligned rows -> always inside tile.
    #pragma unroll
    for (int pe = tid; pe < 512; pe += 256) {
      int m2 = (pe >> 5) * 2;    // 0,2,...,30
      int cc = pe & 31;
      int grow = r0 + m2;
      size_t prow = (size_t)(grow >> 1) * H_DIM + t0 + cc;
      Hout[prow] = sH[m2][cc] + sH[m2 + 1][cc];
      Cout[prow] = sC[m2][cc] + sC[m2 + 1][cc];
    }
  }
}

extern "C" void kernel_launch(void* const* d_in, const int* in_sizes, int n_in,
                              void* d_out, int out_size, void* d_ws, size_t ws_size,
                              hipStream_t stream) {
  (void)in_sizes; (void)n_in; (void)out_size; (void)ws_size;
  const int*   tok  = (const int*)d_in[0];
  // d_in[1] = parent (int64): structure is the implicit complete heap, unused.
  const float* emb  = (const float*)d_in[2];
  const float* Wiou = (const float*)d_in[3];
  const float* biou = (const float*)d_in[4];
  const float* Wf   = (const float*)d_in[5];
  const float* bf   = (const float*)d_in[6];
  float* out = (float*)d_out;

  // Ping-pong hsum/csum buffers in workspace (~96 MiB total).
  char* ws = (char*)d_ws;
  const size_t szA = (size_t)B_DIM * 256 * H_DIM * sizeof(float); // 33.5 MB (parents of level 9)
  const size_t szB = (size_t)B_DIM * 128 * H_DIM * sizeof(float); // 16.8 MB (parents of level 8)
  float* HsA = (float*)(ws);
  float* CsA = (float*)(ws + szA);
  float* HsB = (float*)(ws + 2 * szA);
  float* CsB = (float*)(ws + 2 * szA + szB);

  for (int d = 9; d >= 0; --d) {
    int L = 1 << d;
    int rows = B_DIM * L;
    dim3 grid(rows / 32, H_DIM / 32);
    const float *Hin = nullptr, *Cin = nullptr;
    float *Hout, *Cout;
    int hasState = (d != 9);
    if (hasState) {
      if (d & 1) { Hin = HsB; Cin = CsB; }   // written by even level d+1
      else       { Hin = HsA; Cin = CsA; }   // written by odd  level d+1
    }
    if (d != 0) {
      if (d & 1) { Hout = HsA; Cout = CsA; }
      else       { Hout = HsB; Cout = CsB; }
    } else {
      Hout = HsB; Cout = CsB;                // unused at root
    }
    tree_lstm_level<<<grid, dim3(256), 0, stream>>>(
        tok, emb, Wiou, biou, Wf, bf, Hin, Cin, Hout, Cout, out,
        L, d, L - 1, hasState);
  }
}